// CBOW_60241211293750
// MI455X (gfx1250) — compile-verified
//
#include <hip/hip_runtime.h>
#include <hip/hip_bf16.h>
#include <math.h>

typedef float v2f __attribute__((ext_vector_type(2)));
typedef float v8f __attribute__((ext_vector_type(8)));

#define Bsz    64
#define Ddim   256
#define Vvoc   32000
#define NSLICE 8

// K-split for GEMM1: 32000 K-elems = 8000 k4-steps = 160 chunks * 50 k4-steps
#define KCHUNKS       160
#define K4_PER_CHUNK  50

// workspace layout (float element offsets)
#define X_OFF       0ull                                  // 64*32000
#define LOGITS_OFF  (X_OFF + (size_t)Bsz * Vvoc)          // 64*32000
#define HPART_OFF   (LOGITS_OFF + (size_t)Bsz * Vvoc)     // 160*64*256
#define H_OFF       (HPART_OFF + (size_t)KCHUNKS * Bsz * Ddim)
#define STATS_OFF   (H_OFF + (size_t)Bsz * Ddim)          // 64

static __device__ __forceinline__ v8f wmma_f32_k4(v2f a, v2f b, v8f c) {
  return __builtin_amdgcn_wmma_f32_16x16x4_f32(false, a, false, b, (short)0, c,
                                               false, false);
}

// ---------------------------------------------------------------------------
// 1) x[b,v] = mean over 8 context slices (coalesced streaming)
__global__ void cbow_mean_kernel(const float* __restrict__ in,
                                 float* __restrict__ x) {
  int i = blockIdx.x * blockDim.x + threadIdx.x;
  if (i >= Bsz * Vvoc) return;
  int b = i / Vvoc;
  int v = i - b * Vvoc;
  const float* p = in + (size_t)b * (NSLICE * Vvoc) + v;
  float s = 0.f;
#pragma unroll
  for (int t = 0; t < NSLICE; ++t) s += p[(size_t)t * Vvoc];
  x[i] = s * (1.0f / NSLICE);
}

// ---------------------------------------------------------------------------
// 2) GEMM1 partials: hpart[chunk, b, d] = sum_{v in chunk} x[b,v] * W1[d,v]
//    block = 256 threads = 8 waves; wave w: m_tile = w>>1, n_half = w&1
//    each wave: 16(M) x 128(N) output via 8 accumulators, K-chunk of 200 elems
__global__ __launch_bounds__(256) void cbow_gemm1_kernel(
    const float* __restrict__ x, const float* __restrict__ W1,
    float* __restrict__ hpart) {
  const int lane   = threadIdx.x & 31;
  const int wave   = threadIdx.x >> 5;
  const int m_tile = wave >> 1;          // 0..3  (batch rows m_tile*16..+15)
  const int n_base = (wave & 1) * 128;   // 0 or 128 (D columns)
  const int row    = lane & 15;          // M index / N index within tile
  const int kh     = lane >> 4;          // K-half: lanes 0-15 -> K{0,1}, 16-31 -> K{2,3}
  const int chunk  = blockIdx.x;

  v8f acc[8];
#pragma unroll
  for (int t = 0; t < 8; ++t) {
    v8f z = {0.f, 0.f, 0.f, 0.f, 0.f, 0.f, 0.f, 0.f};
    acc[t] = z;
  }

  const float* aptr = x + (size_t)(m_tile * 16 + row) * Vvoc;
  const float* bptr[8];
#pragma unroll
  for (int t = 0; t < 8; ++t)
    bptr[t] = W1 + (size_t)(n_base + t * 16 + row) * Vvoc;

  const int k0 = chunk * (K4_PER_CHUNK * 4);
  for (int kk = 0; kk < K4_PER_CHUNK; ++kk) {
    const int v = k0 + kk * 4 + kh * 2;
    v2f a = *(const v2f*)(aptr + v);      // A[M=row][v], A[M=row][v+1]
#pragma unroll
    for (int t = 0; t < 8; ++t) {
      v2f bb = *(const v2f*)(bptr[t] + v);  // B[k][N=row] = W1[d][v]
      acc[t] = wmma_f32_k4(a, bb, acc[t]);
    }
  }

  float* out = hpart + (size_t)chunk * (Bsz * Ddim);
#pragma unroll
  for (int t = 0; t < 8; ++t) {
    const int d = n_base + t * 16 + row;
#pragma unroll
    for (int r = 0; r < 8; ++r) {
      const int bo = m_tile * 16 + kh * 8 + r;  // C/D layout: M = r + 8*kh
      out[(size_t)bo * Ddim + d] = acc[t][r];
    }
  }
}

// ---------------------------------------------------------------------------
// 3) h[b,d] = b1[d] + fixed-order sum of partials  (deterministic)
__global__ void cbow_reduce_h_kernel(const float* __restrict__ hpart,
                                     const float* __restrict__ b1,
                                     float* __restrict__ h) {
  int i = blockIdx.x * blockDim.x + threadIdx.x;
  if (i >= Bsz * Ddim) return;
  float s = b1[i & (Ddim - 1)];
  for (int c = 0; c < KCHUNKS; ++c) s += hpart[(size_t)c * (Bsz * Ddim) + i];
  h[i] = s;
}

// ---------------------------------------------------------------------------
// 4) GEMM2: logits[b,v] = sum_d h[b,d] * W2[v,d] + b2[v]
//    block = 128 threads = 4 waves; wave = m_tile; 128 N-columns per block
__global__ __launch_bounds__(128) void cbow_gemm2_kernel(
    const float* __restrict__ h, const float* __restrict__ W2,
    const float* __restrict__ b2, float* __restrict__ logits) {
  const int lane    = threadIdx.x & 31;
  const int m_tile  = threadIdx.x >> 5;  // 0..3
  const int row     = lane & 15;
  const int kh      = lane >> 4;
  const int n_block = blockIdx.x * 128;

  v8f acc[8];
#pragma unroll
  for (int t = 0; t < 8; ++t) {
    v8f z = {0.f, 0.f, 0.f, 0.f, 0.f, 0.f, 0.f, 0.f};
    acc[t] = z;
  }

  const float* aptr = h + (size_t)(m_tile * 16 + row) * Ddim;
  const float* bptr[8];
#pragma unroll
  for (int t = 0; t < 8; ++t)
    bptr[t] = W2 + (size_t)(n_block + t * 16 + row) * Ddim;

  for (int k4 = 0; k4 < Ddim / 4; ++k4) {
    const int d = k4 * 4 + kh * 2;
    v2f a = *(const v2f*)(aptr + d);
#pragma unroll
    for (int t = 0; t < 8; ++t) {
      v2f bb = *(const v2f*)(bptr[t] + d);  // B[k][N] = W2[v][d]
      acc[t] = wmma_f32_k4(a, bb, acc[t]);
    }
  }

#pragma unroll
  for (int t = 0; t < 8; ++t) {
    const int v = n_block + t * 16 + row;
    const float bias = b2[v];
#pragma unroll
    for (int r = 0; r < 8; ++r) {
      const int bo = m_tile * 16 + kh * 8 + r;
      logits[(size_t)bo * Vvoc + v] = acc[t][r] + bias;
    }
  }
}

// ---------------------------------------------------------------------------
// 5) per-row online log-sum-exp: stats[b] = max + log(sum exp(x - max))
__global__ __launch_bounds__(256) void cbow_rowstats_kernel(
    const float* __restrict__ logits, float* __restrict__ stats) {
  __shared__ float sm[256];
  __shared__ float ss[256];
  const int b = blockIdx.x, tid = threadIdx.x;
  const float* rowp = logits + (size_t)b * Vvoc;

  float m = -INFINITY, s = 0.f;
  for (int v = tid; v < Vvoc; v += 256) {
    float xv = rowp[v];
    if (xv > m) {
      s = s * expf(m - xv) + 1.f;
      m = xv;
    } else {
      s += expf(xv - m);
    }
  }
  sm[tid] = m;
  ss[tid] = s;
  __syncthreads();
  for (int off = 128; off > 0; off >>= 1) {
    if (tid < off) {
      float m2 = sm[tid + off], s2 = ss[tid + off];
      float M = fmaxf(sm[tid], m2);
      ss[tid] = ss[tid] * expf(sm[tid] - M) + s2 * expf(m2 - M);
      sm[tid] = M;
    }
    __syncthreads();
  }
  if (tid == 0) stats[b] = sm[0] + logf(ss[0]);
}

// ---------------------------------------------------------------------------
// 6) out = logits - logsumexp(row)
__global__ void cbow_finalize_kernel(const float* __restrict__ logits,
                                     const float* __restrict__ stats,
                                     float* __restrict__ out) {
  int i = blockIdx.x * blockDim.x + threadIdx.x;
  if (i >= Bsz * Vvoc) return;
  int b = i / Vvoc;
  out[i] = logits[i] - stats[b];
}

// ---------------------------------------------------------------------------
extern "C" void kernel_launch(void* const* d_in, const int* in_sizes, int n_in,
                              void* d_out, int out_size, void* d_ws,
                              size_t ws_size, hipStream_t stream) {
  const float* input_vec = (const float*)d_in[0];
  const float* W1 = (const float*)d_in[1];
  const float* b1 = (const float*)d_in[2];
  const float* W2 = (const float*)d_in[3];
  const float* b2 = (const float*)d_in[4];
  float* out = (float*)d_out;
  float* ws = (float*)d_ws;

  float* x      = ws + X_OFF;
  float* logits = ws + LOGITS_OFF;
  float* hpart  = ws + HPART_OFF;
  float* h      = ws + H_OFF;
  float* stats  = ws + STATS_OFF;

  cbow_mean_kernel<<<(Bsz * Vvoc + 255) / 256, 256, 0, stream>>>(input_vec, x);
  cbow_gemm1_kernel<<<KCHUNKS, 256, 0, stream>>>(x, W1, hpart);
  cbow_reduce_h_kernel<<<(Bsz * Ddim + 255) / 256, 256, 0, stream>>>(hpart, b1,
                                                                     h);
  cbow_gemm2_kernel<<<Vvoc / 128, 128, 0, stream>>>(h, W2, b2, logits);
  cbow_rowstats_kernel<<<Bsz, 256, 0, stream>>>(logits, stats);
  cbow_finalize_kernel<<<(Bsz * Vvoc + 255) / 256, 256, 0, stream>>>(logits,
                                                                     stats, out);
}